// InstanceNorm_40209483825523
// MI455X (gfx1250) — compile-verified
//
#include <hip/hip_runtime.h>

// ---------------------------------------------------------------------------
// Per-segment InstanceNorm (PyG semantics), MI455X / gfx1250.
//
// Memory-bound op: 2 passes over x (512MB each) + 512MB store => ~66us floor
// at 23.3 TB/s. Pass 1 computes per-(segment,feature) sum and sum-of-squares
// using the "scatter matmul" formulation on V_WMMA_F32_16X16X4_F32:
//   D(16x16) = A(16x4) x B(4x16) + C,  chained over 4 K-steps = 16 rows.
// A is a 0/1 selection matrix built from the sorted segment ids (exact in
// f32), B is a 16-col tile of x (or x^2). Accumulator rows are flushed to a
// 4MB L2-resident stats buffer with global_atomic_add_f32.
// ---------------------------------------------------------------------------

#define NSEG 4096
#define DFEAT 128
#define EPSV 1e-5f

typedef __attribute__((ext_vector_type(2))) float v2f;
typedef __attribute__((ext_vector_type(4))) float v4f;
typedef __attribute__((ext_vector_type(8))) float v8f;

// workspace layout (floats): sum[NSEG*DFEAT] | sumsq[NSEG*DFEAT] | cnt[NSEG]
#define STAT_FLOATS (2 * NSEG * DFEAT + NSEG)

#define CHUNKS_PER_BLOCK 32
#define ROWS_PER_BLOCK (16 * CHUNKS_PER_BLOCK)

// ---------------------------------------------------------------------------
// Pass 1: segment sum / sum-of-squares via WMMA scatter-matmul.
// Block = 256 threads = 8 waves; wave w owns columns [16w, 16w+16).
// All 8 waves walk the same 16-row chunks (full 512B rows stay coalesced
// across the block). All branches around WMMA are wave-uniform (EXEC = ~0).
// ---------------------------------------------------------------------------
__global__ __launch_bounds__(256) void in_stats_wmma(
    const float* __restrict__ x, const int* __restrict__ batch,
    float* __restrict__ sum, float* __restrict__ sumsq, float* __restrict__ cnt,
    int N)
{
  const int lane  = threadIdx.x & 31;
  const int wave  = threadIdx.x >> 5;
  const int c0    = wave * 16;           // column tile base
  const int ncol  = lane & 15;           // N-index (B/C/D) and M-index (A)
  const int khalf = (lane >> 4) << 1;    // K sub-pair owned by this lane half
  const int rbase = blockIdx.x * ROWS_PER_BLOCK;

  for (int ci = 0; ci < CHUNKS_PER_BLOCK; ++ci) {
    const int r = rbase + ci * 16;       // N % 16 == 0, uniform guard
    if (r >= N) break;

    // --- segment ids of the 16 chunk rows live in lanes 0..15 -------------
    int id = 0;
    if (lane < 16) id = batch[r + lane];
    int idp = __shfl(id, lane > 0 ? lane - 1 : 0);
    int slot = (lane > 0 && lane < 16 && id != idp) ? 1 : 0;
    // inclusive prefix sum over lanes 0..15 -> slot = rank of distinct seg
    #pragma unroll
    for (int off = 1; off < 16; off <<= 1) {
      int v = __shfl(slot, lane - off);
      if (lane >= off) slot += v;
    }
    const int nslots = __shfl(slot, 15) + 1;   // distinct segments in chunk
    if (lane >= 16) slot = 0x7fffffff;         // lanes >=16 hold no row

    // --- scatter matmul: csum += A*X, csq += A*(X.*X) ---------------------
    v8f csum = {};
    v8f csq  = {};
    #pragma unroll
    for (int kc = 0; kc < 4; ++kc) {
      const int kk = (kc << 2) + khalf;        // rows (r+kk, r+kk+1) for lane
      const int s0 = __shfl(slot, kk);
      const int s1 = __shfl(slot, kk + 1);
      v2f a;                                   // A 16x4 f32 layout (ISA 7.12.2)
      a.x = (s0 == ncol) ? 1.0f : 0.0f;
      a.y = (s1 == ncol) ? 1.0f : 0.0f;
      const float* px = x + (size_t)(r + kk) * DFEAT + c0 + ncol;
      v2f b;  b.x  = px[0];       b.y  = px[DFEAT];   // B 4x16: K in VGPR/halves
      v2f b2; b2.x = b.x * b.x;   b2.y = b.y * b.y;
      csum = __builtin_amdgcn_wmma_f32_16x16x4_f32(
          false, a, false, b,  (short)0, csum, false, false);
      csq  = __builtin_amdgcn_wmma_f32_16x16x4_f32(
          false, a, false, b2, (short)0, csq,  false, false);
    }

    // --- flush the nslots live accumulator rows (usually 1) ---------------
    // C/D layout: VGPR j holds row j (lanes 0-15) and row j+8 (lanes 16-31).
    #pragma unroll
    for (int m = 0; m < 16; ++m) {
      if (m < nslots) {                        // wave-uniform predicate
        unsigned long long bm = __ballot(lane < 16 && slot == m);
        unsigned int m16 = (unsigned int)(bm & 0xFFFFull);
        int fl  = __ffs(m16) - 1;
        int seg = __shfl(id, fl);
        float vs = csum[m & 7];
        float vq = csq[m & 7];
        bool act = (m < 8) ? (lane < 16) : (lane >= 16);
        if (act) {
          size_t o = (size_t)seg * DFEAT + c0 + ncol;
          atomicAdd(sum + o,   vs);
          atomicAdd(sumsq + o, vq);
        }
        if (wave == 0 && lane == 0)
          atomicAdd(cnt + seg, (float)__popc(m16));
      }
    }
  }
}

// ---------------------------------------------------------------------------
// Pass 2: streaming apply, one float4 (16B) per thread.
// out = (x - mean) * rsqrt(var + eps) * w + b ; stats stay in L2 (4MB).
// ---------------------------------------------------------------------------
__global__ __launch_bounds__(256) void in_apply(
    const float* __restrict__ x, const int* __restrict__ batch,
    const float* __restrict__ w, const float* __restrict__ bias,
    const float* __restrict__ sum, const float* __restrict__ sumsq,
    const float* __restrict__ cnt, float* __restrict__ out, int N)
{
  const int idx   = blockIdx.x * blockDim.x + threadIdx.x;
  const int total = N * (DFEAT / 4);
  if (idx >= total) return;
  const int r = idx >> 5;            // row
  const int q = (idx & 31) << 2;     // column base (float4 group)

  const int g = batch[r];
  float c = cnt[g];
  if (c < 1.0f) c = 1.0f;            // reference clips counts to >= 1
  const float rc = 1.0f / c;

  const v4f xv = __builtin_nontemporal_load((const v4f*)(x   + (size_t)r * DFEAT + q));
  const v4f sv = *(const v4f*)(sum   + (size_t)g * DFEAT + q);
  const v4f qv = *(const v4f*)(sumsq + (size_t)g * DFEAT + q);
  const v4f wv = *(const v4f*)(w    + q);
  const v4f bv = *(const v4f*)(bias + q);

  v4f o;
  #pragma unroll
  for (int i = 0; i < 4; ++i) {
    float mean = sv[i] * rc;
    float var  = qv[i] * rc - mean * mean;
    float inv  = rsqrtf(var + EPSV);
    o[i] = (xv[i] - mean) * inv * wv[i] + bv[i];
  }
  __builtin_nontemporal_store(o, (v4f*)(out + (size_t)r * DFEAT + q));
}

// ---------------------------------------------------------------------------
extern "C" void kernel_launch(void* const* d_in, const int* in_sizes, int n_in,
                              void* d_out, int out_size, void* d_ws, size_t ws_size,
                              hipStream_t stream)
{
  const float* x     = (const float*)d_in[0];
  const int*   batch = (const int*)d_in[1];
  const float* w     = (const float*)d_in[2];
  const float* b     = (const float*)d_in[3];
  float*       out   = (float*)d_out;
  const int    N     = in_sizes[1];          // batch length = number of rows

  float* sum   = (float*)d_ws;
  float* sumsq = sum + NSEG * DFEAT;
  float* cnt   = sumsq + NSEG * DFEAT;

  // Zero the accumulators every call (atomics below; graph-replay safe).
  hipMemsetAsync(d_ws, 0, (size_t)STAT_FLOATS * sizeof(float), stream);

  const int blocks1 = (N + ROWS_PER_BLOCK - 1) / ROWS_PER_BLOCK;
  in_stats_wmma<<<blocks1, 256, 0, stream>>>(x, batch, sum, sumsq, cnt, N);

  const int total   = N * (DFEAT / 4);
  const int blocks2 = (total + 255) / 256;
  in_apply<<<blocks2, 256, 0, stream>>>(x, batch, w, b, sum, sumsq, cnt, out, N);
}